// KnotEntangle_48180943127161
// MI455X (gfx1250) — compile-verified
//
#include <hip/hip_runtime.h>

// KnotEntangle for MI455X (gfx1250, wave32).
//
// Reference collapses to:
//   r[k]       = sum_s smear[k,s]                       (DC bin of FFT)
//   sumreal[k] = S * smear[k,0]                         (sum of FFT bins)
//   corr[i,j]  = r[i]*r[j]/S ; mixm = gauss(corr; ent params of col), 0 diag
//   a[k] = sum_j mixm[k,j]*cos(pol_j)*sumreal[j] + (K-1) - sum_j mixm[k,j]
//          + sin(pol_k) * sum_i sumreal[i]*mixm[i,k]
//   result[s] = sum_k a[k]*smear[k,s]      <-- f32 matvec -> V_WMMA_F32_16X16X4_F32
//   out[s] = (sum_k gauss_env[k,s]) * result[s]
//
// ws layout (floats): [0,1024) r | [1024,2048) sumreal | [2048,3072) c=cos*sumreal
//                     [3072,4096) sin(pol) | [4096,5120) a

#define KNOTS 1024
#define NSAMP 4096      // SAMPLES
#define NS    4095      // S = SAMPLES-1

typedef __attribute__((ext_vector_type(2))) float v2f;
typedef __attribute__((ext_vector_type(8))) float v8f;

// gauss(x; mean, lo, hi) = exp(-0.5*((x-mean)/exp(sel))^2), sel = x<=mean ? lo : hi.
// 1/exp(sel) == exp(-sel): avoids the IEEE divide chain. __expf -> v_exp_f32
// (quarter-rate TRANS, co-executes with VALU); args are small so fast path is safe.
__device__ __forceinline__ float irr_gauss(float xx, float mean, float lo, float hi) {
    float sel    = (xx <= mean) ? lo : hi;
    float invstd = __expf(-sel);
    float d      = (xx - mean) * invstd;
    return __expf(-0.5f * d * d);
}

__device__ __forceinline__ float wave_sum(float v) {
#pragma unroll
    for (int off = 16; off > 0; off >>= 1) v += __shfl_xor(v, off, 32);
    return v;
}

// ---------------- Phase 1: per-knot stats (one wave per knot) ----------------
__global__ __launch_bounds__(256)
void knot_stats(const float* __restrict__ x, const float* __restrict__ sw,
                const float* __restrict__ km, const float* __restrict__ kl,
                const float* __restrict__ kh, const float* __restrict__ pol,
                float* __restrict__ r, float* __restrict__ sumreal,
                float* __restrict__ ccoef, float* __restrict__ sinp) {
    const int w    = (blockIdx.x * blockDim.x + threadIdx.x) >> 5;  // knot id
    const int lane = threadIdx.x & 31;
    const float lower = sw[0], upper = sw[1];
    const float inv   = 1.0f / (float)NSAMP;

    const float xk    = x[w];
    const float xstep = (upper - lower) * xk * inv;
    const float xlow  = (1.0f - lower) * xk;
    const float m = km[w], lo = kl[w], hi = kh[w];

    float acc = 0.0f;
    for (int s = lane; s < NS; s += 32) {
        float g = xstep * ((float)s * inv) + xlow;
        acc += irr_gauss(g, m, lo, hi);
    }
    acc = wave_sum(acc);
    if (lane == 0) {
        float s0 = irr_gauss(xlow, m, lo, hi);   // smear[k,0] (grid(s=0)=xlow)
        float sr = (float)NS * s0;               // sum of FFT bins = S*smear[k,0]
        r[w]       = acc;
        sumreal[w] = sr;
        float p = pol[w];
        ccoef[w] = __cosf(p) * sr;
        sinp[w]  = __sinf(p);
    }
}

// ------- Phase 2: K x K mix reductions (one wave per row i == col j = w) -----
__global__ __launch_bounds__(256)
void mix_stats(const float* __restrict__ r, const float* __restrict__ sumreal,
               const float* __restrict__ ccoef, const float* __restrict__ sinp,
               const float* __restrict__ em, const float* __restrict__ el,
               const float* __restrict__ eh, float* __restrict__ acoef) {
    const int w    = (blockIdx.x * blockDim.x + threadIdx.x) >> 5;
    const int lane = threadIdx.x & 31;
    const float invS = 1.0f / (float)NS;

    const float ri  = r[w];
    const float emw = em[w], elw = el[w], ehw = eh[w];

    float rowsum = 0.0f, coef = 0.0f, colacc = 0.0f;
#pragma unroll 4
    for (int t = 0; t < KNOTS / 32; ++t) {
        int j = lane + (t << 5);
        if (j != w) {
            float corr = ri * r[j] * invS;                    // symmetric in (i,j)
            float m1 = irr_gauss(corr, em[j], el[j], eh[j]);  // mixm[w][j]
            rowsum += m1;
            coef   += m1 * ccoef[j];
            float m2 = irr_gauss(corr, emw, elw, ehw);        // mixm[j][w]
            colacc += sumreal[j] * m2;
        }
    }
    rowsum = wave_sum(rowsum);
    coef   = wave_sum(coef);
    colacc = wave_sum(colacc);
    if (lane == 0)
        acoef[w] = coef + (float)(KNOTS - 1) - rowsum + sinp[w] * colacc;
}

// --- Phase 3: result[s] = a . smear[:,s] via V_WMMA_F32_16X16X4_F32; ---------
// --- envelope + final product. One wave per 16-sample tile. ------------------
__global__ __launch_bounds__(32)
void fuse(const float* __restrict__ x, const float* __restrict__ sw,
          const float* __restrict__ km, const float* __restrict__ kl,
          const float* __restrict__ kh, const float* __restrict__ acoef,
          float* __restrict__ out, int nout) {
    const int lane = threadIdx.x;        // 0..31, full wave, EXEC all ones
    const int s0   = blockIdx.x * 16;
    const int m    = lane & 15;          // sample row within tile (A M-index)
    const int half = lane >> 4;
    const float lower = sw[0], upper = sw[1];
    const float inv = 1.0f / (float)NSAMP;
    const float xi  = (float)(s0 + m) * inv;   // x_iter for this lane's sample

    // A 16x4 f32 layout (ISA 7.12.2): lanes 0-15: VGPR0=K0, VGPR1=K1;
    //                                 lanes 16-31: VGPR0=K2, VGPR1=K3.
    // B 4x16 broadcast of a[k0..k0+3] across N, same lane-half K mapping.
    const int ka = half << 1;            // 0 or 2
    v8f acc = {};                        // C/D accumulator tile (all cols equal)

    for (int kc = 0; kc < KNOTS; kc += 4) {
        int k0 = kc + ka, k1 = k0 + 1;
        float x0 = x[k0], x1 = x[k1];
        float g0 = ((upper - lower) * x0 * inv) * xi + (1.0f - lower) * x0;
        float g1 = ((upper - lower) * x1 * inv) * xi + (1.0f - lower) * x1;
        v2f a, b;
        a[0] = irr_gauss(g0, km[k0], kl[k0], kh[k0]);  // smear[k0, s0+m]
        a[1] = irr_gauss(g1, km[k1], kl[k1], kh[k1]);  // smear[k1, s0+m]
        b[0] = acoef[k0];
        b[1] = acoef[k1];
        // D[m,n] += sum_kk smear[kc+kk, s0+m] * a_coef[kc+kk]   (n-independent)
        acc = __builtin_amdgcn_wmma_f32_16x16x4_f32(
                  false, a, false, b, (short)0, acc, false, false);
    }

    // Envelope: env[s] = sum_k gauss(g; x[k], lows[k], highs[k]); split k by half.
    float envp = 0.0f;
    for (int k = half; k < KNOTS; k += 2) {
        float xk = x[k];
        float lo = (1.0f - lower) * xk;
        float hi = (1.0f + upper) * xk;
        float g  = (hi - lo) * xi + lo;
        envp += irr_gauss(g, xk, lo, hi);
    }
    envp += __shfl_xor(envp, 16, 32);    // combine the two half-lane partials

    // Extract result tile from C layout: lane0 holds M=0..7 (col 0),
    // lane16 holds M=8..15. Columns are identical so col choice is free.
    __shared__ float res[16];
    __shared__ float env[16];
    if (lane == 0) {
#pragma unroll
        for (int v = 0; v < 8; ++v) res[v] = acc[v];
    }
    if (lane == 16) {
#pragma unroll
        for (int v = 0; v < 8; ++v) res[8 + v] = acc[v];
    }
    if (lane < 16) env[lane] = envp;
    __syncthreads();

    if (lane < 16) {
        int s = s0 + lane;
        if (s < nout) out[s] = env[lane] * res[lane];
    }
}

extern "C" void kernel_launch(void* const* d_in, const int* in_sizes, int n_in,
                              void* d_out, int out_size, void* d_ws, size_t ws_size,
                              hipStream_t stream) {
    const float* x   = (const float*)d_in[0];
    const float* sw  = (const float*)d_in[1];   // smear_window[2]
    const float* km  = (const float*)d_in[2];
    const float* kl  = (const float*)d_in[3];
    const float* kh  = (const float*)d_in[4];
    const float* em  = (const float*)d_in[5];
    const float* el  = (const float*)d_in[6];
    const float* eh  = (const float*)d_in[7];
    const float* pol = (const float*)d_in[8];
    float* out = (float*)d_out;

    float* ws      = (float*)d_ws;
    float* r       = ws;
    float* sumreal = ws + 1024;
    float* ccoef   = ws + 2048;
    float* sinp    = ws + 3072;
    float* acoef   = ws + 4096;

    // 1024 waves = 128 blocks x 256 threads (8 waves/block)
    knot_stats<<<KNOTS / 8, 256, 0, stream>>>(x, sw, km, kl, kh, pol,
                                              r, sumreal, ccoef, sinp);
    mix_stats<<<KNOTS / 8, 256, 0, stream>>>(r, sumreal, ccoef, sinp,
                                             em, el, eh, acoef);
    // 256 tiles x 16 samples covers s = 0..4095 (store guarded to nout=4095)
    fuse<<<NSAMP / 16, 32, 0, stream>>>(x, sw, km, kl, kh, acoef, out, out_size);
}